// ModelAttentionMultiHead_58841051955436
// MI455X (gfx1250) — compile-verified
//
#include <hip/hip_runtime.h>
#include <hip/hip_bf16.h>
#include <math.h>

typedef __attribute__((ext_vector_type(2))) float v2f;
typedef __attribute__((ext_vector_type(8))) float v8f;

#define NNODES 100000
#define NEDGES 1600000
#define NHEADS 5
#define HDIM   12
#define HC     60        // NHEADS*HDIM
#define NC     240       // q|k|v|skip concatenated
#define FEDGE  32
#define NEH    (NEDGES * NHEADS)   // 8,000,000 (exactly 31250*256)
#define INV_SQRT_C 0.28867513459481287f

// ---------------------------------------------------------------------------
// Pack Wq|Wk|Wv|Ws ([K,60] each) into Wcat [K,240], biases into bcat[240]
// ---------------------------------------------------------------------------
__global__ void pack_weights_kernel(const float* __restrict__ Wq, const float* __restrict__ Wk,
                                    const float* __restrict__ Wv, const float* __restrict__ Ws,
                                    const float* __restrict__ bq, const float* __restrict__ bk,
                                    const float* __restrict__ bv, const float* __restrict__ bs,
                                    int K, float* __restrict__ Wcat, float* __restrict__ bcat) {
  int i = blockIdx.x * blockDim.x + threadIdx.x;
  int total = K * NC;
  if (i < total) {
    int k = i / NC, n = i % NC;
    float val;
    if (n < 60)       val = Wq[k * 60 + n];
    else if (n < 120) val = Wk[k * 60 + (n - 60)];
    else if (n < 180) val = Wv[k * 60 + (n - 120)];
    else              val = Ws[k * 60 + (n - 180)];
    Wcat[i] = val;
  }
  if (i < NC) {
    float b;
    if (i < 60)       b = bq[i];
    else if (i < 120) b = bk[i - 60];
    else if (i < 180) b = bv[i - 120];
    else              b = bs[i - 180];
    bcat[i] = b;
  }
}

// ---------------------------------------------------------------------------
// Y[M,240] = X[M,K] @ Wcat[K,240] + bcat   via V_WMMA_F32_16X16X4_F32
// grid = (M/16, 240/16), block = 32 (one wave per 16x16 tile; EXEC all ones)
// ---------------------------------------------------------------------------
__global__ void gemm_qkvs_wmma(const float* __restrict__ X, const float* __restrict__ Wcat,
                               const float* __restrict__ bcat, float* __restrict__ Y, int K) {
  const int mt   = blockIdx.x * 16;
  const int nt   = blockIdx.y * 16;
  const int lane = threadIdx.x;
  const int half = lane >> 4;    // 0: K pair {0,1}; 1: K pair {2,3}
  const int l15  = lane & 15;

  v8f acc = {};
  const float* arow = X    + (size_t)(mt + l15) * K + 2 * half;   // A: row = M, vgprs = K
  const float* bcol = Wcat + (size_t)(2 * half) * NC + nt + l15;  // B: lane = N, vgprs = K

  for (int k0 = 0; k0 < K; k0 += 4) {
    v2f a, b;
    a.x = arow[k0];
    a.y = arow[k0 + 1];
    b.x = bcol[(size_t)k0 * NC];
    b.y = bcol[(size_t)(k0 + 1) * NC];
    acc = __builtin_amdgcn_wmma_f32_16x16x4_f32(false, a, false, b, (short)0, acc, false, false);
  }

  const float bias = bcat[nt + l15];
#pragma unroll
  for (int v = 0; v < 8; ++v) {
    int row = mt + v + 8 * half;   // C/D layout: VGPR v -> M=v (lanes0-15), M=v+8 (lanes16-31)
    Y[(size_t)row * NC + nt + l15] = acc[v] + bias;
  }
}

// ---------------------------------------------------------------------------
// init amax (-inf bits), denom (0), attn accumulator (0)
// ---------------------------------------------------------------------------
__global__ void init_stats_kernel(unsigned int* __restrict__ amax, float* __restrict__ denom,
                                  float* __restrict__ attn) {
  int i = blockIdx.x * blockDim.x + threadIdx.x;
  if (i < NNODES * NHEADS) { amax[i] = 0xFF800000u; denom[i] = 0.0f; }
  if (i < NNODES * HC)     { attn[i] = 0.0f; }
}

// sign-split float atomic max (amax initialized to -inf bits)
__device__ __forceinline__ void atomicMaxFloatBits(unsigned int* addr, float v) {
  if (v >= 0.0f) atomicMax((int*)addr, __float_as_int(v));
  else           atomicMin(addr, __float_as_uint(v));
}

// ---------------------------------------------------------------------------
// Pass 1: alpha[e,h] = <q[dst,h], k[src,h] + e[h]> / sqrt(C); segment max.
// One thread per (edge, head); We staged in LDS; e recomputed inline.
// ---------------------------------------------------------------------------
__global__ void edge_alpha_max_kernel(const int* __restrict__ src, const int* __restrict__ dst,
                                      const float* __restrict__ edge_attr,
                                      const float* __restrict__ We, const float* __restrict__ QKVS,
                                      float* __restrict__ alpha, unsigned int* __restrict__ amax) {
  __shared__ float sWe[FEDGE * HC];   // 7.5 KB
  for (int i = threadIdx.x; i < FEDGE * HC; i += blockDim.x) sWe[i] = We[i];
  __syncthreads();

  unsigned int g = blockIdx.x * blockDim.x + threadIdx.x;
  if (g >= NEH) return;
  const int eid = (int)(g / NHEADS);
  const int h   = (int)(g % NHEADS);
  const int s = src[eid], d = dst[eid];

  const float* ea = edge_attr + (size_t)eid * FEDGE;
  const float* wb = sWe + h * HDIM;

  float ev[HDIM];
#pragma unroll
  for (int c = 0; c < HDIM; ++c) ev[c] = 0.0f;
#pragma unroll 4
  for (int j = 0; j < FEDGE; ++j) {
    const float a = ea[j];
#pragma unroll
    for (int c = 0; c < HDIM; ++c) ev[c] += a * wb[j * HC + c];
  }

  const float* qd = QKVS + (size_t)d * NC + h * HDIM;        // q block
  const float* ks = QKVS + (size_t)s * NC + 60 + h * HDIM;   // k block
  float acc = 0.0f;
#pragma unroll
  for (int c = 0; c < HDIM; ++c) acc += qd[c] * (ks[c] + ev[c]);
  acc *= INV_SQRT_C;

  alpha[g] = acc;
  atomicMaxFloatBits(amax + (size_t)d * NHEADS + h, acc);
}

// ---------------------------------------------------------------------------
// Pass 2: w = exp(alpha - amax[dst]); alpha <- w; denom[dst] += w
// ---------------------------------------------------------------------------
__global__ void edge_softmax_denom_kernel(const int* __restrict__ dst,
                                          const float* __restrict__ amax,
                                          float* __restrict__ alpha_w,
                                          float* __restrict__ denom) {
  unsigned int g = blockIdx.x * blockDim.x + threadIdx.x;
  if (g >= NEH) return;
  const int eid = (int)(g / NHEADS);
  const int h   = (int)(g % NHEADS);
  const int d = dst[eid];
  const float w = expf(alpha_w[g] - amax[(size_t)d * NHEADS + h]);
  alpha_w[g] = w;
  atomicAdd(denom + (size_t)d * NHEADS + h, w);
}

// ---------------------------------------------------------------------------
// Pass 3: attn[dst,h,:] += (w/denom) * (v[src,h,:] + e[h,:])
// ---------------------------------------------------------------------------
__global__ void edge_scatter_kernel(const int* __restrict__ src, const int* __restrict__ dst,
                                    const float* __restrict__ edge_attr,
                                    const float* __restrict__ We, const float* __restrict__ QKVS,
                                    const float* __restrict__ w, const float* __restrict__ denom,
                                    float* __restrict__ attn) {
  __shared__ float sWe[FEDGE * HC];
  for (int i = threadIdx.x; i < FEDGE * HC; i += blockDim.x) sWe[i] = We[i];
  __syncthreads();

  unsigned int g = blockIdx.x * blockDim.x + threadIdx.x;
  if (g >= NEH) return;
  const int eid = (int)(g / NHEADS);
  const int h   = (int)(g % NHEADS);
  const int s = src[eid], d = dst[eid];

  const float coef = w[g] / (denom[(size_t)d * NHEADS + h] + 1e-16f);

  const float* ea = edge_attr + (size_t)eid * FEDGE;
  const float* wb = sWe + h * HDIM;
  float ev[HDIM];
#pragma unroll
  for (int c = 0; c < HDIM; ++c) ev[c] = 0.0f;
#pragma unroll 4
  for (int j = 0; j < FEDGE; ++j) {
    const float a = ea[j];
#pragma unroll
    for (int c = 0; c < HDIM; ++c) ev[c] += a * wb[j * HC + c];
  }

  const float* vs = QKVS + (size_t)s * NC + 120 + h * HDIM;  // v block
  float* out = attn + (size_t)d * HC + h * HDIM;
#pragma unroll
  for (int c = 0; c < HDIM; ++c) atomicAdd(out + c, coef * (vs[c] + ev[c]));
}

// ---------------------------------------------------------------------------
// h = relu(attn + skip), skip = QKVS cols [180,240)
// ---------------------------------------------------------------------------
__global__ void relu_skip_kernel(const float* __restrict__ attn, const float* __restrict__ QKVS,
                                 float* __restrict__ hout) {
  int i = blockIdx.x * blockDim.x + threadIdx.x;
  if (i >= NNODES * HC) return;
  int node = i / HC, j = i % HC;
  float v = attn[i] + QKVS[(size_t)node * NC + 180 + j];
  hout[i] = fmaxf(v, 0.0f);
}

// ---------------------------------------------------------------------------
// Final MLP: out = relu(h @ W1 + b1) @ W2 + b2
// ---------------------------------------------------------------------------
__global__ void mlp_kernel(const float* __restrict__ h, const float* __restrict__ W1,
                           const float* __restrict__ b1, const float* __restrict__ W2,
                           const float* __restrict__ b2, float* __restrict__ out) {
  __shared__ float sW1[60 * 16];
  __shared__ float sB1[16];
  __shared__ float sW2[16 * 2];
  __shared__ float sB2[2];
  for (int i = threadIdx.x; i < 60 * 16; i += blockDim.x) sW1[i] = W1[i];
  if (threadIdx.x < 16) sB1[threadIdx.x] = b1[threadIdx.x];
  if (threadIdx.x < 32) sW2[threadIdx.x] = W2[threadIdx.x];
  if (threadIdx.x < 2)  sB2[threadIdx.x] = b2[threadIdx.x];
  __syncthreads();

  int i = blockIdx.x * blockDim.x + threadIdx.x;
  if (i >= NNODES) return;
  const float* hr = h + (size_t)i * HC;
  float a[16];
#pragma unroll
  for (int t = 0; t < 16; ++t) a[t] = sB1[t];
  for (int j = 0; j < HC; ++j) {
    const float x = hr[j];
#pragma unroll
    for (int t = 0; t < 16; ++t) a[t] += x * sW1[j * 16 + t];
  }
  float o0 = sB2[0], o1 = sB2[1];
#pragma unroll
  for (int t = 0; t < 16; ++t) {
    const float r = fmaxf(a[t], 0.0f);
    o0 += r * sW2[t * 2 + 0];
    o1 += r * sW2[t * 2 + 1];
  }
  out[(size_t)i * 2 + 0] = o0;
  out[(size_t)i * 2 + 1] = o1;
}

// ---------------------------------------------------------------------------
static void run_conv_layer(const float* Xin, int K,
                           const float* Wq, const float* bq, const float* Wk, const float* bk,
                           const float* Wv, const float* bv, const float* We,
                           const float* Ws, const float* bs,
                           const int* src, const int* dst, const float* edge_attr,
                           float* Wcat, float* bcat, float* QKVS, float* alpha,
                           unsigned int* amax, float* denom, float* attn, float* hout,
                           hipStream_t stream) {
  const int packTotal = K * NC;
  pack_weights_kernel<<<(packTotal + 255) / 256, 256, 0, stream>>>(
      Wq, Wk, Wv, Ws, bq, bk, bv, bs, K, Wcat, bcat);
  gemm_qkvs_wmma<<<dim3(NNODES / 16, NC / 16), 32, 0, stream>>>(Xin, Wcat, bcat, QKVS, K);
  init_stats_kernel<<<(NNODES * HC + 255) / 256, 256, 0, stream>>>(amax, denom, attn);
  edge_alpha_max_kernel<<<NEH / 256, 256, 0, stream>>>(src, dst, edge_attr, We, QKVS, alpha, amax);
  edge_softmax_denom_kernel<<<NEH / 256, 256, 0, stream>>>(dst, (const float*)amax, alpha, denom);
  edge_scatter_kernel<<<NEH / 256, 256, 0, stream>>>(src, dst, edge_attr, We, QKVS, alpha, denom, attn);
  relu_skip_kernel<<<(NNODES * HC + 255) / 256, 256, 0, stream>>>(attn, QKVS, hout);
}

extern "C" void kernel_launch(void* const* d_in, const int* in_sizes, int n_in,
                              void* d_out, int out_size, void* d_ws, size_t ws_size,
                              hipStream_t stream) {
  const float* x         = (const float*)d_in[0];
  const int*   edge_idx  = (const int*)d_in[1];
  const float* edge_attr = (const float*)d_in[2];
  const int* src = edge_idx;            // row 0
  const int* dst = edge_idx + NEDGES;   // row 1

  const float* Wq1 = (const float*)d_in[3];  const float* bq1 = (const float*)d_in[4];
  const float* Wk1 = (const float*)d_in[5];  const float* bk1 = (const float*)d_in[6];
  const float* Wv1 = (const float*)d_in[7];  const float* bv1 = (const float*)d_in[8];
  const float* We1 = (const float*)d_in[9];
  const float* Ws1 = (const float*)d_in[10]; const float* bs1 = (const float*)d_in[11];

  const float* Wq2 = (const float*)d_in[12]; const float* bq2 = (const float*)d_in[13];
  const float* Wk2 = (const float*)d_in[14]; const float* bk2 = (const float*)d_in[15];
  const float* Wv2 = (const float*)d_in[16]; const float* bv2 = (const float*)d_in[17];
  const float* We2 = (const float*)d_in[18];
  const float* Ws2 = (const float*)d_in[19]; const float* bs2 = (const float*)d_in[20];

  const float* W1 = (const float*)d_in[21];  const float* b1 = (const float*)d_in[22];
  const float* W2 = (const float*)d_in[23];  const float* b2 = (const float*)d_in[24];

  // workspace layout (floats)
  float* ws = (float*)d_ws;
  size_t o = 0;
  float* Wcat = ws + o;  o += 128 * NC;               // 30720 (max of both layers)
  float* bcat = ws + o;  o += 256;
  float* QKVS = ws + o;  o += (size_t)NNODES * NC;    // 24,000,000
  float* alpha = ws + o; o += (size_t)NEDGES * NHEADS;// 8,000,000
  unsigned int* amax = (unsigned int*)(ws + o); o += (size_t)NNODES * NHEADS;
  float* denom = ws + o; o += (size_t)NNODES * NHEADS;
  float* attn = ws + o;  o += (size_t)NNODES * HC;
  float* h1 = ws + o;    o += (size_t)NNODES * HC;
  float* h2 = ws + o;    o += (size_t)NNODES * HC;

  run_conv_layer(x, 128, Wq1, bq1, Wk1, bk1, Wv1, bv1, We1, Ws1, bs1,
                 src, dst, edge_attr, Wcat, bcat, QKVS, alpha, amax, denom, attn, h1, stream);
  run_conv_layer(h1, 60, Wq2, bq2, Wk2, bk2, Wv2, bv2, We2, Ws2, bs2,
                 src, dst, edge_attr, Wcat, bcat, QKVS, alpha, amax, denom, attn, h2, stream);
  mlp_kernel<<<(NNODES + 127) / 128, 128, 0, stream>>>(h2, W1, b1, W2, b2, (float*)d_out);
}